// SteinRBFkernel_33397665693840
// MI455X (gfx1250) — compile-verified
//
#include <hip/hip_runtime.h>
#include <stdint.h>

#define N_PART 512
#define DIMS   128
#define RANK0  65407u          // (m-1)//2 with m = 512*511/2 = 130816

typedef float v4f __attribute__((ext_vector_type(4)));

// ---------------------------------------------------------------------------
// Kernel 1: per-dimension exact median via 4-pass MSB radix select.
// One block per dimension d. Column p[:,d] staged into LDS with the CDNA5
// async global->LDS copy (ASYNCcnt path), then each pass histograms 8 bits
// of the fp32 bit pattern of (p[j,d]-p[i,d])^2 over all i<j pairs.
// Non-negative floats compare identically to their uint32 bit patterns, so
// the selected 32-bit key is bit-exactly the sorted[RANK0] element.
// ---------------------------------------------------------------------------
__global__ __launch_bounds__(256) void stein_median_kernel(
    const float* __restrict__ p, float* __restrict__ ws) {
  const int d   = blockIdx.x;    // 0..127
  const int tid = threadIdx.x;   // 0..255

  __shared__ float    col[N_PART];
  __shared__ unsigned hist[256];

  // --- Stage column p[:, d] into LDS via async load (gfx1250 path). ---
  for (int i = tid; i < N_PART; i += 256) {
    unsigned lds_byte = (unsigned)(uintptr_t)(&col[i]);   // low 32 bits = LDS addr
    const float* g = p + (size_t)i * DIMS + d;
    asm volatile("global_load_async_to_lds_b32 %0, %1, off"
                 :: "v"(lds_byte), "v"(g)
                 : "memory");
  }
  asm volatile("s_wait_asynccnt 0" ::: "memory");
  __syncthreads();

  unsigned hi_mask = 0u;   // resolved high bits mask
  unsigned hi_val  = 0u;   // resolved high bits value
  unsigned rank    = RANK0;

  for (int pass = 0; pass < 4; ++pass) {
    const int shift = 24 - pass * 8;

    hist[tid] = 0u;
    __syncthreads();

    // All i<j pairs; threads stride over j.
    for (int i = 0; i < N_PART - 1; ++i) {
      const float pi = col[i];
      for (int j = i + 1 + tid; j < N_PART; j += 256) {
        const float    dl  = col[j] - pi;
        const float    v   = dl * dl;
        const unsigned key = __float_as_uint(v);
        if ((key & hi_mask) == hi_val)
          atomicAdd(&hist[(key >> shift) & 255u], 1u);
      }
    }
    __syncthreads();

    // Replicated (deterministic) scan: find bin containing `rank`.
    unsigned cum = 0u;
    unsigned b   = 0u;
    for (; b < 255u; ++b) {
      const unsigned c = hist[b];
      if (cum + c > rank) break;
      cum += c;
    }
    rank   -= cum;
    hi_val |= b << shift;
    hi_mask |= 0xFFu << shift;
    __syncthreads();   // before hist is cleared next pass
  }

  if (tid == 0) {
    const float med = __uint_as_float(hi_val);        // exact median
    const float h   = med / 6.2402758454304058f;      // log(512 + 1)
    const float bw  = fmaxf(h, 1e-9f);
    const float inv = 1.0f / bw;                      // one true divide per dim
    ws[d]        = -inv;          // for log_kernel = norm_sq * (-1/bw)
    ws[DIMS + d] = -2.0f * inv;   // for grad       = delta   * (-2/bw)
  }
}

// ---------------------------------------------------------------------------
// Kernel 2: streaming producer of both 128 MB outputs (256 MB total writes).
// One thread per (pair, d4) float4 unit; consecutive lanes write contiguous
// 512 B. Particle reads are L2-resident (256 KB). Nontemporal stores keep
// the 256 MB one-shot stream from thrashing the 192 MB L2.
// ---------------------------------------------------------------------------
__global__ __launch_bounds__(256) void stein_out_kernel(
    const float* __restrict__ p, const float* __restrict__ ws,
    float* __restrict__ out) {
  const unsigned idx  = blockIdx.x * 256u + threadIdx.x;  // 0 .. 2^23-1
  const unsigned q    = idx & 31u;        // float4 index within the pair row
  const unsigned pair = idx >> 5;         // 0 .. 262143
  const unsigned i    = pair >> 9;        // row
  const unsigned j    = pair & 511u;      // col

  const v4f pj = *reinterpret_cast<const v4f*>(p + j * 128u + q * 4u);
  const v4f pi = *reinterpret_cast<const v4f*>(p + i * 128u + q * 4u);
  const v4f ni = *reinterpret_cast<const v4f*>(ws + q * 4u);         // -1/bw
  const v4f n2 = *reinterpret_cast<const v4f*>(ws + 128u + q * 4u);  // -2/bw

  const v4f dl = pj - pi;
  const v4f lk = (dl * dl) * ni;   // -(norm_sq / bw)
  const v4f gl = dl * n2;          // -2 * delta / bw

  v4f* o0 = reinterpret_cast<v4f*>(out) + (size_t)idx;
  v4f* o1 = reinterpret_cast<v4f*>(out + (size_t)33554432u) + (size_t)idx;
  __builtin_nontemporal_store(lk, o0);
  __builtin_nontemporal_store(gl, o1);
}

// ---------------------------------------------------------------------------
extern "C" void kernel_launch(void* const* d_in, const int* in_sizes, int n_in,
                              void* d_out, int out_size, void* d_ws, size_t ws_size,
                              hipStream_t stream) {
  (void)in_sizes; (void)n_in; (void)out_size; (void)ws_size;
  const float* particles = (const float*)d_in[0];
  float* out = (float*)d_out;
  float* ws  = (float*)d_ws;    // ws[0:128] = -1/bw, ws[128:256] = -2/bw

  stein_median_kernel<<<dim3(DIMS), dim3(256), 0, stream>>>(particles, ws);
  stein_out_kernel<<<dim3(32768), dim3(256), 0, stream>>>(particles, ws, out);
}